// RateRNN_dale_v2_10728828305924
// MI455X (gfx1250) — compile-verified
//
#include <hip/hip_runtime.h>

typedef float v2f __attribute__((ext_vector_type(2)));
typedef float v8f __attribute__((ext_vector_type(8)));
typedef unsigned int u32x4 __attribute__((ext_vector_type(4)));
typedef int i32x4 __attribute__((ext_vector_type(4)));
typedef int i32x8 __attribute__((ext_vector_type(8)));

constexpr int kB = 32, kS = 2048, kIn = 64, kOut = 64, kHid = 512, kExc = 410;
constexpr int kNWG = 16;      // cooperative workgroups in scan kernel
constexpr int kSlice = 32;    // hidden columns per workgroup
constexpr int kKP = 416;      // EXC padded to multiple of 16 for head GEMM
constexpr float kA = 0.1f, kOneMinusA = 0.9f;

__device__ __forceinline__ float sigmoidf_(float v) {
  return 1.0f / (1.0f + __expf(-v));
}

// Monotonic-counter grid barrier across the 16 resident scan workgroups.
// Agent-scope acq/rel gives the L2 visibility + WGP$ invalidate we need for
// the sigmoid(h) exchange buffer.
__device__ __forceinline__ void grid_barrier(unsigned* bar, unsigned phase) {
  __syncthreads();
  if (threadIdx.x == 0) {
    __hip_atomic_fetch_add(bar, 1u, __ATOMIC_ACQ_REL, __HIP_MEMORY_SCOPE_AGENT);
    const unsigned target = (unsigned)kNWG * phase;
    while (__hip_atomic_load(bar, __ATOMIC_ACQUIRE, __HIP_MEMORY_SCOPE_AGENT) < target) {
      __builtin_amdgcn_s_sleep(2);
    }
  }
  __syncthreads();
}

__global__ void barrier_init_kernel(unsigned* bar) {
  if (threadIdx.x == 0) *bar = 0u;
}

// Stage a 2D fp32 tile (rows x 512, row-major, contiguous rows) from global
// memory into LDS with the Tensor Data Mover. Descriptor layout per CDNA5 ISA
// CDNA5 ISA. Issue from ONE wave only (TDM is a per-wave DMA op).
__device__ __forceinline__ void tdm_load_rows_to_lds(const float* gsrc,
                                                     void* lds_dst, int rows) {
  // LDS aperture flat address truncates to the LDS byte address (ISA §10.2).
  unsigned ldsa = (unsigned)(unsigned long long)(uintptr_t)lds_dst;
  unsigned long long ga = (unsigned long long)(uintptr_t)gsrc;

  u32x4 g0;
  g0[0] = 1u;                                    // count=1, user descriptor
  g0[1] = ldsa;                                  // lds_addr
  g0[2] = (unsigned)ga;                          // global_addr[31:0]
  g0[3] = (unsigned)(ga >> 32) | (2u << 30);     // global_addr[56:32], type=2

  i32x8 g1;
  g1[0] = (int)(2u << 16);                       // data_size=4B (code 2)
  g1[1] = (int)(512u << 16);                     // tensor_dim0[15:0]=512
  g1[2] = (int)((unsigned)rows << 16);           // tensor_dim1[15:0]=rows
  g1[3] = (int)(512u << 16);                     // tile_dim0=512
  g1[4] = rows;                                  // tile_dim1=rows, tile_dim2=0
  g1[5] = 512;                                   // tensor_dim0_stride lo32
  g1[6] = 0;
  g1[7] = 0;

  i32x4 gz = {0, 0, 0, 0};
#if __clang_major__ >= 23
  i32x8 gz8 = {0, 0, 0, 0, 0, 0, 0, 0};
  __builtin_amdgcn_tensor_load_to_lds(g0, g1, gz, gz, gz8, 0);
#else
  __builtin_amdgcn_tensor_load_to_lds(g0, g1, gz, gz, 0);
#endif
}

// ---------------------------------------------------------------------------
// Cooperative recurrent scan. 16 WGs x 128 threads (4 waves of 32).
// WG `wg` owns hidden columns [32*wg, 32*wg+32). Its W_hh slab is DMA'd into
// LDS once by the TDM. Per step: 4 waves compute the 32x32 tile of
// sigmoid(h) @ W_hh^T with fp32 WMMA (K=512, 128x V_WMMA_F32_16X16X4_F32),
// fused epilogue applies relu/Dale sign/input current/leak/noise, writes
// hidden to d_out and sigmoid(h_new) to the double-buffered exchange buffer.
// ---------------------------------------------------------------------------
__global__ __launch_bounds__(128)
void rnn_scan_kernel(const float* __restrict__ x,
                     const float* __restrict__ h0,
                     const float* __restrict__ W_ih,
                     const float* __restrict__ W_hh,
                     const float* __restrict__ noise,
                     float* __restrict__ hidden,
                     float* __restrict__ hsig,   // 2 * 32*512 floats in ws
                     unsigned* __restrict__ bar) {
  __shared__ float whh[kSlice * kHid];  // 64 KB: W_hh rows for this slice
  __shared__ float wih[kSlice * kIn];   // 8 KB: W_ih rows (zero for inhibitory)
  __shared__ float ie[kB * kSlice];     // 4 KB: per-step drive (input or osc)

  const int wg = (int)blockIdx.x;
  const int t = (int)threadIdx.x;
  const int wave = t >> 5, lane = t & 31;
  const int hi = lane >> 4, ln = lane & 15;
  const int mi = wave >> 1, ni = wave & 1;  // 2x2 tiles of the 32x32 slice
  const int col0 = wg * kSlice;
  const int mrow = 16 * mi + 8 * hi;        // C-fragment row base for this lane
  const int nloc = 16 * ni + ln;            // local column (0..31)
  const int ncol = col0 + nloc;             // global hidden column
  const float sgn = (ncol < kExc) ? 1.0f : -1.0f;  // Dale sign, fixed per lane

  // Wave 0: TDM DMA of the 32x512 fp32 W_hh slab into LDS (async, TENSORcnt).
  if (wave == 0) {
    tdm_load_rows_to_lds(W_hh + (size_t)col0 * kHid, whh, kSlice);
  }
  // Meanwhile other waves stage W_ih rows (zero for columns >= EXC).
  for (int i = t; i < kSlice * kIn; i += 128) {
    int c = col0 + (i >> 6);
    wih[i] = (c < kExc) ? W_ih[(size_t)c * kIn + (i & 63)] : 0.0f;
  }

  // Recurrent state fragment (C/D layout: vgpr r, lanes0-15 -> M=r, 16-31 -> M=8+r)
  v8f hc;
#pragma unroll
  for (int r = 0; r < 8; ++r) hc[r] = h0[(size_t)(mrow + r) * kHid + ncol];

  // Publish sigmoid(h0) for this slice into exchange buffer 0.
  for (int i = t; i < kB * kSlice; i += 128) {
    int b = i >> 5, c = col0 + (i & 31);
    hsig[(size_t)b * kHid + c] = sigmoidf_(h0[(size_t)b * kHid + c]);
  }
  if (wave == 0) {
    __builtin_amdgcn_s_wait_tensorcnt(0);   // W_hh slab landed in LDS
  }
  grid_barrier(bar, 1u);

  for (int s = 0; s < kS; ++s) {
    const float* hr = hsig + (size_t)(s & 1) * (kB * kHid);        // read buf
    float* hw = hsig + (size_t)((s + 1) & 1) * (kB * kHid);        // write buf
    const float osc_t = __sinf(0.50265482457436692f * (float)s) + 1.0f;

    // Per-step drive for this slice: excitatory columns get
    // relu(x[b,s,:] . W_ih[c,:]); inhibitory columns get the oscillator.
    // Storing both in `ie` makes the hot epilogue fully branchless.
    if (col0 < kExc) {
      const int b = t >> 2, cg = (t & 3) * 8;
      const float* xr = x + ((size_t)b * kS + s) * kIn;
      float acc[8];
#pragma unroll
      for (int j = 0; j < 8; ++j) acc[j] = 0.0f;
      for (int k = 0; k < kIn; k += 4) {
        float4 xv = *(const float4*)(xr + k);
#pragma unroll
        for (int j = 0; j < 8; ++j) {
          const float* wr = &wih[(cg + j) * kIn + k];
          acc[j] += xv.x * wr[0] + xv.y * wr[1] + xv.z * wr[2] + xv.w * wr[3];
        }
      }
#pragma unroll
      for (int j = 0; j < 8; ++j) {
        ie[b * kSlice + cg + j] =
            (col0 + cg + j < kExc) ? fmaxf(acc[j], 0.0f) : osc_t;
      }
      if (s + 1 < kS) __builtin_prefetch(xr + kIn, 0, 0);  // next step's x row
    } else {
      for (int i = t; i < kB * kSlice; i += 128) ie[i] = osc_t;
    }
    __syncthreads();

    // Prefetch this step's noise rows (cold HBM stream, consumed in the
    // epilogue) so the ~128 WMMA K-loop hides the HBM round trip.
#pragma unroll
    for (int r = 0; r < 8; ++r) {
      __builtin_prefetch(noise + ((size_t)s * kB + (mrow + r)) * kHid + ncol, 0, 0);
    }

    // rec = sigmoid(h) @ W_hh^T tile via fp32 WMMA 16x16x4, K = 512.
    // A[m][k]: lane m=ln, vgpr0/1 = k = kb+2*hi (+1)  (contiguous float2)
    // B[k][n] = W_hh[n][k]: lane n=ln, same k split    (contiguous float2)
    v8f c = {};
    const float* arow = hr + (size_t)(16 * mi + ln) * kHid + 2 * hi;
    const float* brow = whh + (size_t)nloc * kHid + 2 * hi;
#pragma unroll 8
    for (int kb = 0; kb < kHid; kb += 4) {
      v2f av = *(const v2f*)(arow + kb);
      v2f bv = *(const v2f*)(brow + kb);
      c = __builtin_amdgcn_wmma_f32_16x16x4_f32(false, av, false, bv, (short)0,
                                                c, false, false);
    }

    // Fully branchless epilogue: I = sgn*relu(rec) + drive, leak + noise.
#pragma unroll
    for (int r = 0; r < 8; ++r) {
      const int m = mrow + r;
      float rec = fmaxf(c[r], 0.0f);
      float I = fmaf(sgn, rec, ie[m * kSlice + nloc]);
      float nz = noise[((size_t)s * kB + m) * kHid + ncol];
      float hn = kOneMinusA * hc[r] + kA * I + 0.01f * nz;
      hc[r] = hn;
      hidden[((size_t)m * kS + s) * kHid + ncol] = hn;
      hw[(size_t)m * kHid + ncol] = sigmoidf_(hn);
    }
    grid_barrier(bar, (unsigned)(s + 2));
  }
}

// ---------------------------------------------------------------------------
// Output head: logits = sigmoid(hidden[:, :, :410]) @ W_ho^T, softmax over 64.
// 512 WGs x 256 threads; each wave owns 16 rows x 64 cols (4 fp32 WMMA accs).
// ---------------------------------------------------------------------------
__global__ __launch_bounds__(256)
void head_kernel(const float* __restrict__ hidden,
                 const float* __restrict__ W_ho,
                 float* __restrict__ out) {
  __shared__ float who[kOut * kKP];  // 104 KB, K zero-padded to 416
  const int t = (int)threadIdx.x;
  for (int i = t; i < kOut * kKP; i += 256) {
    int o = i / kKP, e = i % kKP;
    who[i] = (e < kExc) ? W_ho[(size_t)o * kExc + e] : 0.0f;
  }
  __syncthreads();

  const int wave = t >> 5, lane = t & 31;
  const int hi = lane >> 4, ln = lane & 15;
  const size_t row0 = (size_t)blockIdx.x * 128 + (size_t)wave * 16;

  v8f acc[4] = {};

  const float* arow = hidden + (row0 + (size_t)ln) * kHid + 2 * hi;
#pragma unroll 4
  for (int kb = 0; kb < kKP; kb += 4) {
    v2f av;
    av.x = sigmoidf_(arow[kb]);       // k >= 410 multiplied by zero-padded B
    av.y = sigmoidf_(arow[kb + 1]);
#pragma unroll
    for (int nt = 0; nt < 4; ++nt) {
      v2f bv = *(const v2f*)(&who[(nt * 16 + ln) * kKP + kb + 2 * hi]);
      acc[nt] = __builtin_amdgcn_wmma_f32_16x16x4_f32(false, av, false, bv,
                                                      (short)0, acc[nt],
                                                      false, false);
    }
  }

  // Softmax over the 64 logits of each row: values live as 4 per lane across
  // a 16-lane group (same hi). Reduce with width-16 xor shuffles.
#pragma unroll
  for (int r = 0; r < 8; ++r) {
    float v0 = acc[0][r], v1 = acc[1][r], v2 = acc[2][r], v3 = acc[3][r];
    float mx = fmaxf(fmaxf(v0, v1), fmaxf(v2, v3));
#pragma unroll
    for (int msk = 1; msk < 16; msk <<= 1)
      mx = fmaxf(mx, __shfl_xor(mx, msk, 16));
    float e0 = __expf(v0 - mx), e1 = __expf(v1 - mx);
    float e2 = __expf(v2 - mx), e3 = __expf(v3 - mx);
    float sm = e0 + e1 + e2 + e3;
#pragma unroll
    for (int msk = 1; msk < 16; msk <<= 1)
      sm += __shfl_xor(sm, msk, 16);
    float inv = 1.0f / sm;
    const size_t row = row0 + (size_t)(r + 8 * hi);
    float* orow = out + row * kOut + ln;
    orow[0]  = e0 * inv;
    orow[16] = e1 * inv;
    orow[32] = e2 * inv;
    orow[48] = e3 * inv;
  }
}

extern "C" void kernel_launch(void* const* d_in, const int* in_sizes, int n_in,
                              void* d_out, int out_size, void* d_ws, size_t ws_size,
                              hipStream_t stream) {
  (void)in_sizes; (void)n_in; (void)out_size; (void)ws_size;
  const float* x     = (const float*)d_in[0];
  const float* h0    = (const float*)d_in[1];
  const float* W_ih  = (const float*)d_in[2];
  const float* W_hh  = (const float*)d_in[3];
  const float* W_ho  = (const float*)d_in[4];
  const float* noise = (const float*)d_in[5];

  float* out = (float*)d_out;                                  // (B,S,64)
  float* hidden = out + (size_t)kB * kS * kOut;                // (B,S,512)

  unsigned* bar = (unsigned*)d_ws;                             // barrier counter
  float* hsig = (float*)((char*)d_ws + 256);                   // 2 x 32x512 f32

  barrier_init_kernel<<<1, 32, 0, stream>>>(bar);
  rnn_scan_kernel<<<kNWG, 128, 0, stream>>>(x, h0, W_ih, W_hh, noise,
                                            hidden, hsig, bar);
  head_kernel<<<(kB * kS) / 128, 256, 0, stream>>>(hidden, W_ho, out);
}